// PatchCRPS_25177098289694
// MI455X (gfx1250) — compile-verified
//
#include <hip/hip_runtime.h>
#include <hip/hip_bf16.h>

// ---------------------------------------------------------------------------
// PatchCRPS loss for (8,1,256,256) fp32 inputs, K=9, pad=4.
//   dist = x - avgpool9(x, count_include_pad=False)
//   loss = mean over (B, H*W, 81) of (sort(patch(pred_dist)) - sort(patch(targ_dist)))^2
// ---------------------------------------------------------------------------

#define BATCH   8
#define HS      256
#define WS      256
#define HW      (HS * WS)          // 65536
#define BHW     (BATCH * HW)       // 524288
#define PATCH   81
#define PADR    4
#define FLT_BIG 3.402823466e38f
// total element count for the mean: 8 * 65536 * 81
#define INV_N   (1.0 / 42467328.0)

typedef __attribute__((ext_vector_type(2))) float v2f;
typedef __attribute__((ext_vector_type(8))) float v8f;

// ---------------------------------------------------------------------------
// Kernel 1: horizontal 9-tap box sum as a banded GEMM using
// V_WMMA_F32_16X16X4_F32.  One wave (32 threads) computes one 16x16 tile:
//   D[m][n] = sum_k X[r0+m][k] * band(k, c0+n),  band(k,j)=1 iff |k-j|<=4
// 24 contributing input columns (c0-4 .. c0+19) swept in 6 K=4 steps.
// ---------------------------------------------------------------------------
__global__ void rowsum_wmma_kernel(const float* __restrict__ pred,
                                   const float* __restrict__ targ,
                                   float* __restrict__ rs) {
    int wid    = blockIdx.x;          // 0 .. 4095
    int tensor = wid >> 11;           // 0 = pred, 1 = targ
    int rem    = wid & 2047;
    int img    = rem >> 8;            // 0 .. 7
    int t      = rem & 255;
    int r0     = (t >> 4) << 4;       // tile row base
    int c0     = (t & 15) << 4;       // tile col base

    const float* src = (tensor ? targ : pred) + (size_t)img * HW;
    float*       dst = rs + (size_t)tensor * BHW + (size_t)img * HW;

    int lane = threadIdx.x & 31;
    int half = lane >> 4;
    int lid  = lane & 15;

    v8f acc = {};
#pragma unroll
    for (int s = 0; s < 6; ++s) {
        int k0 = c0 - PADR + 4 * s;   // first of 4 input columns this step
        int ka = k0 + half * 2;       // this lane-half's K columns
        int row = r0 + lid;

        v2f a, b;
        a.x = (ka     >= 0 && ka     < WS) ? src[row * WS + ka]     : 0.0f;
        a.y = (ka + 1 >= 0 && ka + 1 < WS) ? src[row * WS + ka + 1] : 0.0f;

        int j  = c0 + lid;            // output column for this lane
        int d0 = ka - j;
        int d1 = ka + 1 - j;
        b.x = (d0 >= -PADR && d0 <= PADR) ? 1.0f : 0.0f;
        b.y = (d1 >= -PADR && d1 <= PADR) ? 1.0f : 0.0f;

        acc = __builtin_amdgcn_wmma_f32_16x16x4_f32(
            /*neg_a=*/false, a, /*neg_b=*/false, b,
            /*c_mod=*/(short)0, acc, /*reuse_a=*/false, /*reuse_b=*/false);
    }

#pragma unroll
    for (int r = 0; r < 8; ++r) {
        int row = r0 + r + half * 8;
        dst[row * WS + c0 + lid] = acc[r];
    }
}

// ---------------------------------------------------------------------------
// Kernel 2: vertical 9-tap on the row sums, divide by the valid-cell count
// (count_include_pad=False => cnt = cnty * cntx), then dist = x - avg.
// ---------------------------------------------------------------------------
__global__ void dist_kernel(const float* __restrict__ pred,
                            const float* __restrict__ targ,
                            const float* __restrict__ rs,
                            float* __restrict__ dist) {
    int t = blockIdx.x * blockDim.x + threadIdx.x;   // < 2*BHW
    int tensor = (t >= BHW) ? 1 : 0;
    int p = t - tensor * BHW;
    int y = (p >> 8) & 255;
    int x = p & 255;

    const float* r = rs + (size_t)tensor * BHW + p;
    float s = 0.0f;
#pragma unroll
    for (int dy = -PADR; dy <= PADR; ++dy) {
        int yy = y + dy;
        if (yy >= 0 && yy < HS) s += r[dy * WS];
    }
    int cy = min(y + PADR, HS - 1) - max(y - PADR, 0) + 1;
    int cx = min(x + PADR, WS - 1) - max(x - PADR, 0) + 1;
    float srcv = (tensor ? targ : pred)[p];
    dist[t] = srcv - s / (float)(cy * cx);
}

// ---------------------------------------------------------------------------
// Fully unrolled register-resident bitonic sort (ascending), N = power of 2.
// Directions resolve statically after unroll -> pure v_min/v_max pairs.
// ---------------------------------------------------------------------------
template <int N>
__device__ __forceinline__ void bitonic(float (&v)[N]) {
#pragma unroll
    for (int k = 2; k <= N; k <<= 1) {
#pragma unroll
        for (int j = k >> 1; j > 0; j >>= 1) {
#pragma unroll
            for (int i = 0; i < N; ++i) {
                int ixj = i ^ j;
                if (ixj > i) {
                    bool up = ((i & k) == 0);
                    float a = v[i], b = v[ixj];
                    float lo = fminf(a, b), hi = fmaxf(a, b);
                    v[i]   = up ? lo : hi;
                    v[ixj] = up ? hi : lo;
                }
            }
        }
    }
}

// Load patch elements [n0, n1) (linear index n over the 9x9 window, zero
// padded outside the image) into out[0 .. n1-n0).
template <int N0, int N1>
__device__ __forceinline__ void load_range(const float* __restrict__ d,
                                           int y, int x, float* out) {
#pragma unroll
    for (int n = N0; n < N1; ++n) {
        int dy = n / 9 - PADR;        // compile-time after unroll
        int dx = n % 9 - PADR;
        int yy = y + dy, xx = x + dx;
        out[n - N0] = (yy >= 0 && yy < HS && xx >= 0 && xx < WS)
                          ? d[yy * WS + xx] : 0.0f;
    }
}

// ---------------------------------------------------------------------------
// Kernel 3: per pixel (one thread each, one wave per block):
//   - sort each 81-elem patch as a 64-run + 32-run (17 real + 15 FLT_BIG pads)
//     in REGISTERS (peak ~64 live floats), dump runs to conflict-free LDS
//   - single rolled 4-way merge over ranks 0..80 popping the min head of
//     pred's runs and targ's runs, accumulating (pv - tv)^2 in rank order.
// LDS layout: element e of thread t at sh[e*32 + t]  (bank = t or t+32, so
// divergent merge pointers across lanes still hit distinct banks).
// ---------------------------------------------------------------------------
#define SLOT_PA 0     // pred run A (64)
#define SLOT_PB 64    // pred run B (32)
#define SLOT_TA 96    // targ run A (64)
#define SLOT_TB 160   // targ run B (32)

__global__ void sort_mse_kernel(const float* __restrict__ dist,
                                float* __restrict__ partial) {
    __shared__ float sh[192 * 32];
    int tid = threadIdx.x;              // 0..31
    int p   = blockIdx.x * 32 + tid;    // pixel id < BHW
    int img = p >> 16;
    int y   = (p >> 8) & 255;
    int x   = p & 255;

    const float* dp = dist + (size_t)img * HW;           // pred dist
    const float* dt = dist + (size_t)BHW + (size_t)img * HW;  // targ dist

    {
        float v[64];
        load_range<0, 64>(dp, y, x, v);
        bitonic<64>(v);
#pragma unroll
        for (int i = 0; i < 64; ++i) sh[(SLOT_PA + i) * 32 + tid] = v[i];
    }
    {
        float w[32];
        load_range<64, 81>(dp, y, x, w);
#pragma unroll
        for (int i = 17; i < 32; ++i) w[i] = FLT_BIG;
        bitonic<32>(w);
#pragma unroll
        for (int i = 0; i < 32; ++i) sh[(SLOT_PB + i) * 32 + tid] = w[i];
    }
    {
        float v[64];
        load_range<0, 64>(dt, y, x, v);
        bitonic<64>(v);
#pragma unroll
        for (int i = 0; i < 64; ++i) sh[(SLOT_TA + i) * 32 + tid] = v[i];
    }
    {
        float w[32];
        load_range<64, 81>(dt, y, x, w);
#pragma unroll
        for (int i = 17; i < 32; ++i) w[i] = FLT_BIG;
        bitonic<32>(w);
#pragma unroll
        for (int i = 0; i < 32; ++i) sh[(SLOT_TB + i) * 32 + tid] = w[i];
    }

    // rank-synchronized 4-way merge + squared-diff accumulation (branchless)
    int ia = 0, ib = 0, ic = 0, id = 0;
    float acc = 0.0f;
#pragma unroll 1
    for (int i = 0; i < PATCH; ++i) {
        float a = (ia < 64) ? sh[(SLOT_PA + ia) * 32 + tid] : FLT_BIG;
        float b = (ib < 32) ? sh[(SLOT_PB + ib) * 32 + tid] : FLT_BIG;
        float c = (ic < 64) ? sh[(SLOT_TA + ic) * 32 + tid] : FLT_BIG;
        float d = (id < 32) ? sh[(SLOT_TB + id) * 32 + tid] : FLT_BIG;
        bool ta = (a <= b);
        bool tc = (c <= d);
        float pv = ta ? a : b;
        float tv = tc ? c : d;
        ia += ta ? 1 : 0;  ib += ta ? 0 : 1;
        ic += tc ? 1 : 0;  id += tc ? 0 : 1;
        float df = pv - tv;
        acc += df * df;
    }

    // deterministic wave32 reduction
#pragma unroll
    for (int off = 16; off > 0; off >>= 1)
        acc += __shfl_xor(acc, off, 32);
    if (tid == 0) partial[blockIdx.x] = acc;
}

// ---------------------------------------------------------------------------
// Kernel 4: deterministic final reduction of the 16384 block partials.
// ---------------------------------------------------------------------------
__global__ void reduce_kernel(const float* __restrict__ partial,
                              float* __restrict__ out) {
    __shared__ double red[256];
    int tid = threadIdx.x;
    double s = 0.0;
    for (int i = tid; i < BHW / 32; i += 256) s += (double)partial[i];
    red[tid] = s;
    __syncthreads();
    for (int off = 128; off > 0; off >>= 1) {
        if (tid < off) red[tid] += red[tid + off];
        __syncthreads();
    }
    if (tid == 0) out[0] = (float)(red[0] * INV_N);
}

// ---------------------------------------------------------------------------
extern "C" void kernel_launch(void* const* d_in, const int* in_sizes, int n_in,
                              void* d_out, int out_size, void* d_ws, size_t ws_size,
                              hipStream_t stream) {
    const float* pred = (const float*)d_in[0];
    const float* targ = (const float*)d_in[1];

    float* ws      = (float*)d_ws;
    float* rs      = ws;                      // 2*BHW floats (row sums)
    float* dist    = ws + 2 * (size_t)BHW;    // 2*BHW floats
    float* partial = ws;                      // aliases rs (dead by kernel 3)

    // 1) horizontal box sums via WMMA: one wave per 16x16 tile
    rowsum_wmma_kernel<<<2 * BATCH * (HS / 16) * (WS / 16), 32, 0, stream>>>(
        pred, targ, rs);

    // 2) vertical pass + count division + dist
    dist_kernel<<<(2 * BHW) / 256, 256, 0, stream>>>(pred, targ, rs, dist);

    // 3) per-pixel run sorts + rank-merge MSE -> per-block partials
    sort_mse_kernel<<<BHW / 32, 32, 0, stream>>>(dist, partial);

    // 4) deterministic final reduction
    reduce_kernel<<<1, 256, 0, stream>>>(partial, (float*)d_out);
}